// RecurrentNeuralNetwork_36137854829023
// MI455X (gfx1250) — compile-verified
//
#include <hip/hip_runtime.h>

typedef float v2f __attribute__((ext_vector_type(2)));
typedef float v8f __attribute__((ext_vector_type(8)));

#define T_LEN 16384
#define D_DIM 1024
#define H_DIM 8

// ---------------------------------------------------------------------------
// Phase 1: xwT[n][t] = sum_d x[t][d] * W_ih[n][d] + b_ih[n] + b_hh[n]
// One wave computes a 16-timestep x 16-col (8 valid) tile via
// V_WMMA_F32_16X16X4_F32, sweeping K=1024 four at a time. HBM-bound (~64MB).
// Output is TRANSPOSED so the serial scan streams contiguously per lane.
// ---------------------------------------------------------------------------
__global__ __launch_bounds__(256) void rnn_inproj_wmma(
    const float* __restrict__ x, const float* __restrict__ W_ih,
    const float* __restrict__ b_ih, const float* __restrict__ b_hh,
    float* __restrict__ xwT)
{
    const int lane = threadIdx.x & 31;
    const int wave = threadIdx.x >> 5;
    const int tile = blockIdx.x * 8 + wave;    // 16 timesteps per tile
    const int r    = lane & 15;                // A: row within tile / B: col n
    const int hi   = lane >> 4;                // half-wave select
    const int kb   = hi * 2;                   // K sub-offset within frag
    const int nn   = r & 7;                    // clamped h index (pad 8->16)
    const float nmask = (r < 8) ? 1.0f : 0.0f; // zero padded B cols, no EXEC div

    const float* xrow = x    + (size_t)(tile * 16 + r) * D_DIM;
    const float* wrow = W_ih + (size_t)nn * D_DIM;

    v8f acc = {};
#pragma unroll 4
    for (int k = 0; k < D_DIM; k += 4) {
        v2f a = *(const v2f*)(xrow + k + kb);     // A: 16x4 f32 fragment
        v2f b = *(const v2f*)(wrow + k + kb);     // B: 4x16 f32 fragment
        b.x *= nmask;
        b.y *= nmask;
        acc = __builtin_amdgcn_wmma_f32_16x16x4_f32(
            /*neg_a=*/false, a, /*neg_b=*/false, b,
            /*c_mod=*/(short)0, acc, /*reuse_a=*/false, /*reuse_b=*/false);
    }

    if (r < 8) {
        const float bias = b_ih[r] + b_hh[r];
#pragma unroll
        for (int v = 0; v < 8; ++v) {
            const int tl = tile * 16 + v + hi * 8;     // C/D row = timestep
            xwT[(size_t)r * T_LEN + tl] = acc[v] + bias;
        }
    }
}

// ---------------------------------------------------------------------------
// Phase 2: strictly serial ReLU recurrence, one wave.
// Lane i holds h[i] and W_hh[i][:]. Per step: readlane-broadcast h[j],
// 8-deep fmac chain, relu. xwT streamed with a register double buffer so
// memory latency never enters the dependent chain.
// ---------------------------------------------------------------------------
__global__ void rnn_scan(const float* __restrict__ xwT,
                         const float* __restrict__ W_hh,
                         float* __restrict__ hs)
{
    const int lane = threadIdx.x & 31;
    const int i = lane & 7;                     // lanes 8..31 replicate 0..7

    float w[8];
#pragma unroll
    for (int j = 0; j < 8; ++j) w[j] = W_hh[i * 8 + j];

    const float4* xs = (const float4*)(xwT + (size_t)i * T_LEN);
    float4 cur = xs[0];
    float h = 0.0f;

    for (int t = 0; t < T_LEN; t += 4) {
        const int tn = (t + 4 < T_LEN) ? (t + 4) : 0;  // prefetch next block
        float4 nxt = xs[tn >> 2];
        const float xv[4] = {cur.x, cur.y, cur.z, cur.w};
#pragma unroll
        for (int c = 0; c < 4; ++c) {
            float s = xv[c];
#pragma unroll
            for (int j = 0; j < 8; ++j) {
                const float hj = __int_as_float(
                    __builtin_amdgcn_readlane(__float_as_int(h), j));
                s = fmaf(w[j], hj, s);
            }
            h = fmaxf(s, 0.0f);
            if (lane < 8)                        // non-blocking history store
                hs[(size_t)(t + c) * 8 + lane] = h;
        }
        cur = nxt;
    }
}

// ---------------------------------------------------------------------------
// Phase 3: out[t] = hs[t] . W_out + b_out   (parallel, L2-resident)
// ---------------------------------------------------------------------------
__global__ __launch_bounds__(256) void rnn_out(
    const float* __restrict__ hs, const float* __restrict__ W_out,
    const float* __restrict__ b_out, float* __restrict__ out)
{
    const int t = blockIdx.x * blockDim.x + threadIdx.x;
    if (t >= T_LEN) return;
    const float4* hp = (const float4*)(hs + (size_t)t * 8);
    const float4 h0 = hp[0], h1 = hp[1];
    float s = b_out[0];
    s += h0.x * W_out[0] + h0.y * W_out[1] + h0.z * W_out[2] + h0.w * W_out[3];
    s += h1.x * W_out[4] + h1.y * W_out[5] + h1.z * W_out[6] + h1.w * W_out[7];
    out[t] = s;
}

extern "C" void kernel_launch(void* const* d_in, const int* in_sizes, int n_in,
                              void* d_out, int out_size, void* d_ws, size_t ws_size,
                              hipStream_t stream) {
    const float* x     = (const float*)d_in[0];
    const float* W_ih  = (const float*)d_in[1];
    const float* b_ih  = (const float*)d_in[2];
    const float* W_hh  = (const float*)d_in[3];
    const float* b_hh  = (const float*)d_in[4];
    const float* W_out = (const float*)d_in[5];
    const float* b_out = (const float*)d_in[6];
    float* out = (float*)d_out;

    float* xwT = (float*)d_ws;                        // [H][T]  512 KB
    float* hs  = xwT + (size_t)T_LEN * H_DIM;         // [T][H]  512 KB

    // 1024 tiles of 16 timesteps, 8 waves (tiles) per 256-thread block
    rnn_inproj_wmma<<<(T_LEN / 16) / 8, 256, 0, stream>>>(x, W_ih, b_ih, b_hh, xwT);
    rnn_scan<<<1, 32, 0, stream>>>(xwT, W_hh, hs);
    rnn_out<<<(T_LEN + 255) / 256, 256, 0, stream>>>(hs, W_out, b_out, out);
}